// TensorProductEncoder_70016556860030
// MI455X (gfx1250) — compile-verified
//
#include <hip/hip_runtime.h>

typedef float v8f __attribute__((ext_vector_type(8)));
typedef float v2f __attribute__((ext_vector_type(2)));

#if defined(__HIP_DEVICE_COMPILE__)
#if !__has_builtin(__builtin_amdgcn_wmma_f32_16x16x4_f32)
#error "__builtin_amdgcn_wmma_f32_16x16x4_f32 not available on device toolchain"
#endif
#endif

#define D1    256
#define D2    256
#define OUTD  1024
#define BB    64
#define SS    128

#define KTOT   (D1 * D2)        // 65536
#define KSPLIT 32
#define KBLK   (KTOT / KSPLIT)  // 2048 per k-partition
#define KC     64               // k-chunk staged in LDS
#define LDSS   68               // padded LDS row stride (floats): conflict-free b64 reads, 272B rows (16B aligned)

// ---- CDNA5 async global->LDS copies (ASYNCcnt-tracked, VGPR-bypassing) ----
__device__ __forceinline__ void async_ld_b128(void* lds, const void* g)
{
    asm volatile("global_load_async_to_lds_b128 %0, %1, off"
                 :: "v"((unsigned)(unsigned long long)lds),
                    "v"((unsigned long long)g)
                 : "memory");
}
__device__ __forceinline__ void async_ld_b128_nt(void* lds, const void* g)
{
    asm volatile("global_load_async_to_lds_b128 %0, %1, off th:TH_LOAD_NT"
                 :: "v"((unsigned)(unsigned long long)lds),
                    "v"((unsigned long long)g)
                 : "memory");
}

// ---------------------------------------------------------------------------
// Kernel 1: tp[b] = f_b^T @ r_b   (gather fused, fp32 WMMA 16x16x4)
// grid = (16 d-tiles, 64 batches), block = 256 (8 waves); wave handles 2 e-tiles
// ---------------------------------------------------------------------------
__global__ __launch_bounds__(256) void tp_kernel(const int* __restrict__ fillers,
                                                 const int* __restrict__ roles,
                                                 const float* __restrict__ ftab,
                                                 const float* __restrict__ rtab,
                                                 float* __restrict__ tp)
{
    __shared__ int sF[SS];
    __shared__ int sR[SS];

    const int b  = blockIdx.y;
    const int d0 = blockIdx.x * 16;
    const int t  = threadIdx.x;

    if (t < SS)            sF[t]      = fillers[b * SS + t];
    else if (t < 2 * SS)   sR[t - SS] = roles[b * SS + (t - SS)];
    __syncthreads();

    const int wave = t >> 5;
    const int lane = t & 31;
    const int m    = lane & 15;   // A: row M; B: col N
    const int hi   = lane >> 4;   // K half: VGPR0 holds K=2*hi, VGPR1 holds K=2*hi+1

    const int e0a = (wave * 2 + 0) * 16;
    const int e0b = (wave * 2 + 1) * 16;

    v8f acc0 = {};
    v8f acc1 = {};

    for (int s0 = 0; s0 < SS; s0 += 4) {
        const int sA  = s0 + 2 * hi;
        const int if0 = sF[sA], if1 = sF[sA + 1];
        const int ir0 = sR[sA], ir1 = sR[sA + 1];

        v2f a, b0, b1;
        a.x  = ftab[(size_t)if0 * D1 + d0 + m];   // A[m, 2hi]
        a.y  = ftab[(size_t)if1 * D1 + d0 + m];   // A[m, 2hi+1]
        b0.x = rtab[(size_t)ir0 * D2 + e0a + m];  // B[2hi,   n]
        b0.y = rtab[(size_t)ir1 * D2 + e0a + m];  // B[2hi+1, n]
        b1.x = rtab[(size_t)ir0 * D2 + e0b + m];
        b1.y = rtab[(size_t)ir1 * D2 + e0b + m];

        acc0 = __builtin_amdgcn_wmma_f32_16x16x4_f32(false, a, false, b0, (short)0, acc0, false, false);
        acc1 = __builtin_amdgcn_wmma_f32_16x16x4_f32(false, a, false, b1, (short)0, acc1, false, false);
    }

    float* tb = tp + (size_t)b * KTOT;
#pragma unroll
    for (int v = 0; v < 8; ++v) {
        const int row = d0 + v + 8 * hi;          // C layout: VGPR v -> rows v / v+8
        tb[row * D2 + e0a + m] = acc0[v];
        tb[row * D2 + e0b + m] = acc1[v];
    }
}

// ---------------------------------------------------------------------------
// Kernel 2: split-K GEMM  partial[kb] = tp(64 x Kblk) @ W(64n x Kblk)^T
// grid = (16 n-blocks of 64, 32 k-partitions), block = 256 (8 waves)
// Double-buffered LDS slabs filled with global_load_async_to_lds_b128;
// W uses TH_LOAD_NT (read-once stream) so tp stays resident in L2.
// ---------------------------------------------------------------------------
__global__ __launch_bounds__(256) void gemm_kernel(const float* __restrict__ tp,
                                                   const float* __restrict__ W,
                                                   float* __restrict__ partial)
{
    __shared__ float sT[2][64 * LDSS];
    __shared__ float sW[2][64 * LDSS];

    const int n0    = blockIdx.x * 64;
    const int kb    = blockIdx.y;
    const int kbase = kb * KBLK;

    const int t    = threadIdx.x;
    const int q    = t & 15;    // float4 column within slab row
    const int r0   = t >> 4;    // base slab row (0..15)
    const int wave = t >> 5;
    const int lane = t & 31;
    const int m    = lane & 15;
    const int hi   = lane >> 4;

    const int mt  = wave >> 1;         // 0..3 : m-tile (batch rows)
    const int nt0 = (wave & 1) * 2;    // 0..3 : n-tiles (two per wave)
    const int nt1 = nt0 + 1;

    v8f acc0 = {};
    v8f acc1 = {};

    // async fill of one (tp,W) slab pair: 8 async-load instructions per wave
    auto fill = [&](int buf, int kc) {
#pragma unroll
        for (int rr = 0; rr < 64; rr += 16) {
            const int row = r0 + rr;
            async_ld_b128(&sT[buf][row * LDSS + q * 4],
                          &tp[(size_t)row * KTOT + kbase + kc + q * 4]);
            async_ld_b128_nt(&sW[buf][row * LDSS + q * 4],
                             &W[(size_t)(n0 + row) * KTOT + kbase + kc + q * 4]);
        }
    };

    fill(0, 0);

    for (int kc = 0, ic = 0; kc < KBLK; kc += KC, ++ic) {
        const int cur = ic & 1;

        if (kc + KC < KBLK) {
            fill(cur ^ 1, kc + KC);                        // prefetch next chunk
            asm volatile("s_wait_asynccnt 0x8" ::: "memory"); // in-order: current chunk's 8 done
        } else {
            asm volatile("s_wait_asynccnt 0x0" ::: "memory");
        }
        __syncthreads();   // all waves' fills for buffer `cur` visible

#pragma unroll
        for (int kk = 0; kk < KC; kk += 4) {
            const v2f a  = *(const v2f*)&sT[cur][(mt * 16 + m) * LDSS + kk + 2 * hi];
            const v2f b0 = *(const v2f*)&sW[cur][(nt0 * 16 + m) * LDSS + kk + 2 * hi];
            const v2f b1 = *(const v2f*)&sW[cur][(nt1 * 16 + m) * LDSS + kk + 2 * hi];
            acc0 = __builtin_amdgcn_wmma_f32_16x16x4_f32(false, a, false, b0, (short)0, acc0, false, false);
            acc1 = __builtin_amdgcn_wmma_f32_16x16x4_f32(false, a, false, b1, (short)0, acc1, false, false);
        }
        __syncthreads();   // compute done before this buffer is refilled
    }

    float* p = partial + (size_t)kb * (BB * OUTD);
#pragma unroll
    for (int v = 0; v < 8; ++v) {
        const int row = mt * 16 + v + 8 * hi;   // batch row 0..63
        p[row * OUTD + n0 + nt0 * 16 + m] = acc0[v];
        p[row * OUTD + n0 + nt1 * 16 + m] = acc1[v];
    }
}

// ---------------------------------------------------------------------------
// Kernel 3: out = sum_kb partial[kb] + bias   (fixed order -> deterministic)
// ---------------------------------------------------------------------------
__global__ __launch_bounds__(256) void reduce_kernel(const float* __restrict__ partial,
                                                     const float* __restrict__ bias,
                                                     float* __restrict__ out)
{
    const int idx = blockIdx.x * 256 + threadIdx.x;   // 0..65535
    const int n   = idx & (OUTD - 1);
    float s = bias[n];
#pragma unroll 8
    for (int kb = 0; kb < KSPLIT; ++kb)
        s += partial[(size_t)kb * (BB * OUTD) + idx];
    out[idx] = s;
}

// ---------------------------------------------------------------------------
extern "C" void kernel_launch(void* const* d_in, const int* in_sizes, int n_in,
                              void* d_out, int out_size, void* d_ws, size_t ws_size,
                              hipStream_t stream)
{
    const int*   fillers = (const int*)d_in[0];
    const int*   roles   = (const int*)d_in[1];
    const float* ftab    = (const float*)d_in[2];
    const float* rtab    = (const float*)d_in[3];
    const float* W       = (const float*)d_in[4];
    const float* bias    = (const float*)d_in[5];
    float*       out     = (float*)d_out;

    float* tp      = (float*)d_ws;                 // 64*65536 f32 = 16 MB
    float* partial = tp + (size_t)BB * KTOT;       // 32*64*1024 f32 = 8 MB

    tp_kernel<<<dim3(16, BB), 256, 0, stream>>>(fillers, roles, ftab, rtab, tp);
    gemm_kernel<<<dim3(OUTD / 64, KSPLIT), 256, 0, stream>>>(tp, W, partial);
    reduce_kernel<<<dim3(BB * OUTD / 256), 256, 0, stream>>>(partial, bias, out);
}